// Sparsemax_68607807586695
// MI455X (gfx1250) — compile-verified
//
#include <hip/hip_runtime.h>
#include <hip/hip_bf16.h>
#include <stdint.h>

// Row-wise masked sparsemax, N=16384 rows x D=2048 cols, f32.
// Memory-bound: 384 MiB traffic @ 23.3 TB/s ~ 17 us floor.
//  - one 256-thread block (8 x wave32) per row
//  - async bulk DMA of the row into LDS (global_load_async_to_lds_b128 with
//    TH_LOAD_NT: read-once stream, don't pollute the 192MB L2), ASYNCcnt wait
//  - 8 elements/thread held in registers; bisection on the simplex threshold
//    theta in [zmax-1, zmax] (26 iters), exact refine over the support
//  - ping-pong LDS reduction buffers -> ONE barrier per block reduction
//  - nontemporal b128 stores for the write-once output

#define DCOLS 2048
#define THREADS 256
#define EPT 8

typedef float v4f __attribute__((ext_vector_type(4)));

__device__ __forceinline__ void async_ld_b128_nt(uint32_t lds_off, uint64_t gaddr) {
  // VDST = LDS byte address (low 32 bits of flat shared pointer),
  // VADDR = 64-bit global address, saddr = off.  Tracked by ASYNCcnt.
  // NT temporal hint: row is consumed exactly once.
  asm volatile("global_load_async_to_lds_b128 %0, %1, off th:TH_LOAD_NT"
               :: "v"(lds_off), "v"(gaddr)
               : "memory");
}

__device__ __forceinline__ void wait_asynccnt0() {
  asm volatile("s_wait_asynccnt 0" ::: "memory");
}

// One barrier per reduction: caller alternates `red` between two buffers.
// Safe because a buffer written in call k is only re-written in call k+2,
// and every thread passed call (k+1)'s barrier -- which happens-after all
// call-k reads -- before that.
__device__ __forceinline__ float block_sum(float v, volatile float* red, int tid) {
  #pragma unroll
  for (int off = 16; off > 0; off >>= 1)
    v += __shfl_xor(v, off, 32);            // wave32 tree
  if ((tid & 31) == 0) red[tid >> 5] = v;
  __syncthreads();
  float s = 0.0f;
  #pragma unroll
  for (int i = 0; i < 8; ++i) s += red[i];  // identical order in every lane
  return s;
}

__device__ __forceinline__ float block_max(float v, volatile float* red, int tid) {
  #pragma unroll
  for (int off = 16; off > 0; off >>= 1)
    v = fmaxf(v, __shfl_xor(v, off, 32));
  if ((tid & 31) == 0) red[tid >> 5] = v;
  __syncthreads();
  float s = -INFINITY;
  #pragma unroll
  for (int i = 0; i < 8; ++i) s = fmaxf(s, red[i]);
  return s;
}

__global__ void __launch_bounds__(THREADS)
sparsemax_kernel(const float* __restrict__ x, const int* __restrict__ mask,
                 float* __restrict__ out) {
  __shared__ __align__(16) float lds_x[DCOLS];
  __shared__ __align__(16) int   lds_m[DCOLS];
  __shared__ float red[2][8];               // ping-pong reduction buffers

  const int row = blockIdx.x;
  const int tid = threadIdx.x;

  const float* xrow = x    + (size_t)row * DCOLS;
  const int*   mrow = mask + (size_t)row * DCOLS;

  // ---- async DMA row into LDS: 4 x b128 per thread covers 2 x 8KB ----
  const uint32_t lx = (uint32_t)(uintptr_t)&lds_x[0];
  const uint32_t lm = (uint32_t)(uintptr_t)&lds_m[0];
  const uint64_t gx = (uint64_t)(uintptr_t)xrow;
  const uint64_t gm = (uint64_t)(uintptr_t)mrow;
  const uint32_t o0 = (uint32_t)tid * 16u;          // bytes 0..4095
  const uint32_t o1 = 4096u + (uint32_t)tid * 16u;  // bytes 4096..8191
  async_ld_b128_nt(lx + o0, gx + o0);
  async_ld_b128_nt(lx + o1, gx + o1);
  async_ld_b128_nt(lm + o0, gm + o0);
  async_ld_b128_nt(lm + o1, gm + o1);
  wait_asynccnt0();        // this wave's async writes to LDS have landed
  __syncthreads();         // make them visible workgroup-wide

  // ---- pull this thread's 8 elements into registers ----
  // element ids: e0..e0+3 = 4*tid, e1..e1+3 = 1024 + 4*tid
  const int e0 = tid * 4;
  const int e1 = 1024 + tid * 4;
  float z[EPT];
  float lcnt = 0.0f;
  float lmax = -INFINITY;
  #pragma unroll
  for (int j = 0; j < 4; ++j) {
    const bool a0 = lds_m[e0 + j] != 0;
    const bool a1 = lds_m[e1 + j] != 0;
    z[j]     = a0 ? lds_x[e0 + j] : -INFINITY;
    z[4 + j] = a1 ? lds_x[e1 + j] : -INFINITY;
    lcnt += (a0 ? 1.0f : 0.0f) + (a1 ? 1.0f : 0.0f);
    lmax  = fmaxf(lmax, fmaxf(z[j], z[4 + j]));
  }

  int ph = 0;  // ping-pong phase (uniform control flow across the block)
  const float nact = block_sum(lcnt, red[ph], tid); ph ^= 1;
  const float zmax = block_max(lmax, red[ph], tid); ph ^= 1;

  float theta = 0.0f;
  if (nact > 0.0f) {
    // f(theta) = sum max(z - theta, 0); f(zmax-1) >= 1 >= f(zmax) = 0
    float lo = zmax - 1.0f;
    float hi = zmax;
    #pragma unroll 1
    for (int it = 0; it < 26; ++it) {
      const float mid = 0.5f * (lo + hi);
      float p = 0.0f;
      #pragma unroll
      for (int j = 0; j < EPT; ++j) p += fmaxf(z[j] - mid, 0.0f);
      const float f = block_sum(p, red[ph], tid); ph ^= 1;
      if (f >= 1.0f) lo = mid; else hi = mid;   // uniform across block
    }
    // exact refine over support S = { z > lo } (a tie exactly at theta leaves
    // theta invariant, so this equals the reference's (csum_rho - 1)/rho)
    float ps = 0.0f, pc = 0.0f;
    #pragma unroll
    for (int j = 0; j < EPT; ++j) {
      if (z[j] > lo) { ps += z[j]; pc += 1.0f; }
    }
    const float ssum = block_sum(ps, red[ph], tid); ph ^= 1;
    const float scnt = block_sum(pc, red[ph], tid); ph ^= 1;
    theta = (ssum - 1.0f) / fmaxf(scnt, 1.0f);
  }

  // ---- write p = max(z - theta, 0); masked lanes have z = -inf -> 0 ----
  float* orow = out + (size_t)row * DCOLS;
  v4f r0, r1;
  #pragma unroll
  for (int j = 0; j < 4; ++j) {
    r0[j] = fmaxf(z[j]     - theta, 0.0f);
    r1[j] = fmaxf(z[4 + j] - theta, 0.0f);
  }
  __builtin_nontemporal_store(r0, (v4f*)(orow + e0));
  __builtin_nontemporal_store(r1, (v4f*)(orow + e1));
}

extern "C" void kernel_launch(void* const* d_in, const int* in_sizes, int n_in,
                              void* d_out, int out_size, void* d_ws, size_t ws_size,
                              hipStream_t stream) {
  const float* x    = (const float*)d_in[0];
  const int*   mask = (const int*)d_in[1];
  float*       out  = (float*)d_out;
  const int rows = out_size / DCOLS;   // 16384
  hipLaunchKernelGGL(sparsemax_kernel, dim3(rows), dim3(THREADS), 0, stream,
                     x, mask, out);
}